// ChamferDistance_11261404250604
// MI455X (gfx1250) — compile-verified
//
#include <hip/hip_runtime.h>

typedef __attribute__((ext_vector_type(2))) float v2f;
typedef __attribute__((ext_vector_type(4))) float v4f;
typedef __attribute__((ext_vector_type(8))) float v8f;

#define NB 4
#define P1 8192
#define P2 8192

// ---------------------------------------------------------------------------
// Pre-pass: build augmented target array  aug[i] = {y0, y1, y2, |y|^2}
// (i = batch*P2 + point). 16B per point -> A-operand becomes one b64 load.
// ---------------------------------------------------------------------------
__global__ void chamfer_aug_kernel(const float* __restrict__ tgt,
                                   v4f* __restrict__ aug, int total) {
    int i = blockIdx.x * blockDim.x + threadIdx.x;
    if (i < total) {
        float y0 = tgt[i * 3 + 0];
        float y1 = tgt[i * 3 + 1];
        float y2 = tgt[i * 3 + 2];
        v4f v;
        v.x = y0; v.y = y1; v.z = y2;
        v.w = fmaf(y0, y0, fmaf(y1, y1, y2 * y2));
        aug[i] = v;
    }
}

__global__ void chamfer_zero_kernel(float* out) {
    if (threadIdx.x == 0 && blockIdx.x == 0) out[0] = 0.0f;
}

// ---------------------------------------------------------------------------
// Main kernel: one wave32 owns 16 source points; scans all P2 targets in
// 16-wide tiles via V_WMMA_F32_16X16X4_F32 with the distance fully folded in:
//   A[m][k] = {y0,y1,y2,|y|^2},  B[k][n] = {-2x0,-2x1,-2x2,1}
//   D[m][n] = |y_m|^2 - 2 x_n . y_m          (add |x_n|^2 after the argmin)
// ---------------------------------------------------------------------------
__global__ void __launch_bounds__(256)
chamfer_nn_kernel(const float* __restrict__ src,
                  const float* __restrict__ aug,  // v4f per target point
                  float* __restrict__ out) {
    const int lane = threadIdx.x & 31;
    const int wid  = blockIdx.x * (blockDim.x >> 5) + (threadIdx.x >> 5);
    const int batch = wid >> 9;          // 512 source tiles per batch
    const int stile = wid & 511;
    const int l15   = lane & 15;
    const bool hi   = (lane >= 16);

    // ---- B operand: source n = stile*16 + l15 --------------------------
    const int nsrc = stile * 16 + l15;
    const float* sp = src + ((size_t)batch * P1 + (size_t)nsrc) * 3;
    const float x0 = sp[0], x1 = sp[1], x2 = sp[2];
    const float xn = fmaf(x0, x0, fmaf(x1, x1, x2 * x2));
    v2f b;
    b.x = hi ? (-2.0f * x2) : (-2.0f * x0);   // K2 : K0
    b.y = hi ? 1.0f         : (-2.0f * x1);   // K3 : K1

    // ---- scan target tiles ---------------------------------------------
    // lane reads {y0,y1} (lo half) or {y2,|y|^2} (hi half) of target l15
    const float* abase = aug + ((size_t)batch * P2) * 4 + (hi ? 2 : 0);
    float runmin = 3.402823466e+38f;

    #pragma unroll 4
    for (int t = 0; t < P2 / 16; ++t) {
        const v2f a = *(const v2f*)(abase + (size_t)(t * 16 + l15) * 4);
        v8f c = {};
        // (neg_a, A, neg_b, B, c_mod, C, reuse_a, reuse_b)
        v8f d = __builtin_amdgcn_wmma_f32_16x16x4_f32(
            false, a, false, b, (short)0, c, false, false);
        float m = fminf(fminf(fminf(d[0], d[1]), fminf(d[2], d[3])),
                        fminf(fminf(d[4], d[5]), fminf(d[6], d[7])));
        runmin = fminf(runmin, m);
    }

    // merge targets m=0..7 (lo lanes) with m=8..15 (hi lanes)
    runmin = fminf(runmin, __shfl_xor(runmin, 16, 32));

    // nn squared distance for source n, clamped like the reference
    float val = fmaxf(runmin + xn, 0.0f);
    if (hi) val = 0.0f;                       // count each source once

    // wave-wide sum of this wave's 16 source contributions
    #pragma unroll
    for (int off = 1; off < 32; off <<= 1)
        val += __shfl_xor(val, off, 32);

    if (lane == 0)
        atomicAdd(out, val * (1.0f / (float)NB));  // batch_reduction='mean'
}

// ---------------------------------------------------------------------------
extern "C" void kernel_launch(void* const* d_in, const int* in_sizes, int n_in,
                              void* d_out, int out_size, void* d_ws, size_t ws_size,
                              hipStream_t stream) {
    const float* src = (const float*)d_in[0];   // (4, 8192, 3) f32
    const float* tgt = (const float*)d_in[1];   // (4, 8192, 3) f32
    float* out = (float*)d_out;                 // scalar f32
    v4f*   aug = (v4f*)d_ws;                    // NB*P2*16B = 512 KB scratch

    chamfer_zero_kernel<<<1, 32, 0, stream>>>(out);

    const int total_tgt = NB * P2;
    chamfer_aug_kernel<<<(total_tgt + 255) / 256, 256, 0, stream>>>(tgt, aug, total_tgt);

    const int waves  = NB * (P1 / 16);          // 2048 waves
    const int blocks = waves / 8;               // 256 threads = 8 waves/block
    chamfer_nn_kernel<<<blocks, 256, 0, stream>>>(src, (const float*)aug, out);
}